// GCN_with_CRF_59442347377127
// MI455X (gfx1250) — compile-verified
//
#include <hip/hip_runtime.h>

// ---------------------------------------------------------------------------
// GCN_with_CRF on MI455X (gfx1250, wave32, WMMA)
//
// Math note: group_softmax with seg = arange(n) has singleton groups, so
// x_crf == 1.0 exactly. The CRF scatter is dead code; h2 = 0.9*h + 0.1.
// Pipeline: h = relu(gcn1(x)); h2 = 0.9h+0.1; out = log_softmax(gcn2(h2)).
// GEMMs use V_WMMA_F32_16X16X4_F32 (exact f32 matrix pipe); edge scatter is
// the real cost and is L2-resident (hw fits in 192MB L2).
// ---------------------------------------------------------------------------

typedef float v2f __attribute__((ext_vector_type(2)));
typedef float v8f __attribute__((ext_vector_type(8)));

#define WAVE 32

// ---------------------------------------------------------------- zero fill
__global__ __launch_bounds__(256) void zero_kernel(float* __restrict__ p, size_t count) {
    size_t i = (size_t)blockIdx.x * blockDim.x + threadIdx.x;
    size_t stride = (size_t)gridDim.x * blockDim.x;
    for (; i < count; i += stride) p[i] = 0.0f;
}

// ------------------------------------------------------- degree accumulation
__global__ __launch_bounds__(256) void degree_kernel(const int* __restrict__ dst,
                                                     float* __restrict__ deg, int E) {
    int i = blockIdx.x * blockDim.x + threadIdx.x;
    if (i < E) atomicAdd(&deg[dst[i]], 1.0f);
}

// deg -> dinv = rsqrt(deg + 1)   (self loop adds 1)
__global__ __launch_bounds__(256) void rsqrt_kernel(float* __restrict__ deg, int n) {
    int i = blockIdx.x * blockDim.x + threadIdx.x;
    if (i < n) deg[i] = rsqrtf(deg[i] + 1.0f);
}

// ------------------------------------------------------- WMMA f32 GEMM
// Y[N x M] = X[N x K] @ W[K x M], row-major, no bias (bias fused later).
// Block: 256 threads = 8 waves, each wave computes a 16 x M stripe.
// A-fragment layout (V_WMMA_F32_16X16X4_F32, 16x4):
//   v0: lanes 0-15 -> K=k+0, lanes 16-31 -> K=k+2 ; v1: K=k+1 / K=k+3
// B-fragment (4x16): v0: lanes0-15 -> row k+0 col=lane, lanes16-31 -> row k+2
// C/D (16x16): vgpr i holds row (i + 8*(lane>>4)), col = lane&15.
//
// LDS W layout: K-group-of-4 interleaved, element (k,n) at dword
//   (k/4)*(4*M) + n*4 + (k%4)
// so each lane's B pair {W[k+koff][n], W[k+koff+1][n]} is one aligned
// ds_load_b64 (pairs merge into ds_load_2addr_b64) and the wave touches all
// 64 banks once (lanes 0-15: dwords 4n+{0,1}; lanes 16-31: 4n+{2,3}).
// B fragments of a k-step are loaded into an array BEFORE the wmma group so
// the ds loads issue back-to-back (MLP) instead of load->wait->wmma bubbles.
template <int K, int M>
__global__ __launch_bounds__(256) void gemm_wmma_kernel(const float* __restrict__ X,
                                                        const float* __restrict__ W,
                                                        float* __restrict__ Y,
                                                        int nrows) {
    constexpr int KCHUNK = 64;
    constexpr int NT = M / 16;
    static_assert(K % KCHUNK == 0, "K chunking");
    __shared__ float Wlds[KCHUNK * M];   // 32KB (M=128) / 16KB (M=64)

    const int tid  = threadIdx.x;
    const int lane = tid & (WAVE - 1);
    const int wave = tid >> 5;
    const int rowBase = blockIdx.x * 128 + wave * 16;

    // A-fragment addressing for this lane
    int arow = rowBase + (lane & 15);
    if (arow >= nrows) arow = nrows - 1;          // clamp: EXEC must stay all-1 for WMMA
    const int koff  = (lane >> 4) * 2;            // 0 for lanes 0-15, 2 for 16-31
    const int nlane = lane & 15;
    const int bln   = nlane * 4 + koff;           // lane's dword offset within a K-group row

    v8f acc[NT];
#pragma unroll
    for (int nt = 0; nt < NT; ++nt) acc[nt] = (v8f)(0.0f);

    for (int kc = 0; kc < K; kc += KCHUNK) {
        // cooperative load of W chunk into LDS, K-group-of-4 interleaved
        for (int idx = tid; idx < KCHUNK * M; idx += 256) {
            int kk = idx / M;
            int n  = idx - kk * M;
            Wlds[(kk >> 2) * (4 * M) + n * 4 + (kk & 3)] = W[(size_t)kc * M + idx];
        }
        __syncthreads();

        // prefetch next W chunk (one 128B line per thread) -> global_prefetch_b8
        if (kc + KCHUNK < K) {
            if (tid < (KCHUNK * M) / 32)
                __builtin_prefetch(W + (size_t)(kc + KCHUNK) * M + (size_t)tid * 32, 0, 1);
        }

        const float* xptr = X + (size_t)arow * K + kc + koff;
#pragma unroll
        for (int kk = 0; kk < KCHUNK; kk += 4) {
            v2f a = *(const v2f*)(xptr + kk);
            const float* brow = &Wlds[(kk >> 2) * (4 * M) + bln];
            // 1) issue all B-fragment loads for this k-step
            v2f b[NT];
#pragma unroll
            for (int nt = 0; nt < NT; ++nt) b[nt] = *(const v2f*)(brow + nt * 64);
            // 2) then the wmma group
#pragma unroll
            for (int nt = 0; nt < NT; ++nt) {
                acc[nt] = __builtin_amdgcn_wmma_f32_16x16x4_f32(
                    /*neg_a=*/false, a, /*neg_b=*/false, b[nt],
                    /*c_mod=*/(short)0, acc[nt], /*reuse_a=*/false, /*reuse_b=*/false);
            }
        }
        __syncthreads();
    }

    // store D: vgpr i -> row rowBase + i + 8*(lane>>4), col nt*16 + (lane&15)
    const int rhalf = (lane >> 4) * 8;
#pragma unroll
    for (int nt = 0; nt < NT; ++nt) {
#pragma unroll
        for (int i = 0; i < 8; ++i) {
            int r = rowBase + rhalf + i;
            if (r < nrows) Y[(size_t)r * M + nt * 16 + nlane] = acc[nt][i];
        }
    }
}

// ------------------------------------------------------- edge scatter
// agg[dst] += dinv[src]*dinv[dst] * h[src], one wave per edge, F/32 floats/lane
template <int F>
__global__ __launch_bounds__(256) void scatter_kernel(const int* __restrict__ src,
                                                      const int* __restrict__ dst,
                                                      const float* __restrict__ dinv,
                                                      const float* __restrict__ h,
                                                      float* __restrict__ agg, int E) {
    constexpr int V = F / WAVE;   // 4 (F=128) or 2 (F=64)
    const int lane = threadIdx.x & (WAVE - 1);
    const int wave = threadIdx.x >> 5;
    const int e = blockIdx.x * 8 + wave;
    if (e >= E) return;
    const int s = src[e];
    const int d = dst[e];
    const float c = dinv[s] * dinv[d];
    const float* hp = h + (size_t)s * F + lane * V;
    float* ap = agg + (size_t)d * F + lane * V;
    float vals[V];
#pragma unroll
    for (int j = 0; j < V; ++j) vals[j] = hp[j];   // merges to b128/b64 load
#pragma unroll
    for (int j = 0; j < V; ++j) atomicAdd(ap + j, c * vals[j]);
}

// ------------------------------------------------------- conv1 epilogue
// h2 = 0.9*relu(agg + dinv^2*hw + b1) + 0.1   (0.1 = ALPHA * x_crf, x_crf==1)
__global__ __launch_bounds__(256) void finalize1_kernel(float* __restrict__ agg_h2,
                                                        const float* __restrict__ hw,
                                                        const float* __restrict__ dinv,
                                                        const float* __restrict__ b1,
                                                        int total) {
    int idx = blockIdx.x * blockDim.x + threadIdx.x;
    if (idx >= total) return;
    int row = idx >> 7;          // F = 128
    int f = idx & 127;
    float di = dinv[row];
    float v = agg_h2[idx] + di * di * hw[idx] + b1[f];
    v = fmaxf(v, 0.0f);
    agg_h2[idx] = 0.9f * v + 0.1f;
}

// ------------------------------------------------------- conv2 epilogue + log_softmax
// out_row = v - max(v) - log(sum exp(v - max(v))), v = agg2 + dinv^2*hw2 + b2
// One wave per row (OUT=64 -> 2 cols per lane), wave32 shuffle reductions.
__global__ __launch_bounds__(256) void finalize2_kernel(const float* __restrict__ agg2,
                                                        const float* __restrict__ hw2,
                                                        const float* __restrict__ dinv,
                                                        const float* __restrict__ b2,
                                                        float* __restrict__ out, int n) {
    const int lane = threadIdx.x & (WAVE - 1);
    const int wave = threadIdx.x >> 5;
    const int row = blockIdx.x * 8 + wave;
    if (row >= n) return;
    const float di = dinv[row];
    const float di2 = di * di;
    const size_t base = (size_t)row * 64;
    float v0 = agg2[base + lane] + di2 * hw2[base + lane] + b2[lane];
    float v1 = agg2[base + 32 + lane] + di2 * hw2[base + 32 + lane] + b2[32 + lane];
    float m = fmaxf(v0, v1);
#pragma unroll
    for (int o = 16; o > 0; o >>= 1) m = fmaxf(m, __shfl_xor(m, o, WAVE));
    float s = __expf(v0 - m) + __expf(v1 - m);
#pragma unroll
    for (int o = 16; o > 0; o >>= 1) s += __shfl_xor(s, o, WAVE);
    float l = m + __logf(s);
    out[base + lane] = v0 - l;
    out[base + 32 + lane] = v1 - l;
}

// ---------------------------------------------------------------------------
extern "C" void kernel_launch(void* const* d_in, const int* in_sizes, int n_in,
                              void* d_out, int out_size, void* d_ws, size_t ws_size,
                              hipStream_t stream) {
    constexpr int IN = 256, HID = 128, OUT = 64;
    const float* x  = (const float*)d_in[0];
    const int*   ei = (const int*)d_in[1];
    // d_in[2] = edge_weight: mathematically unused (CRF term collapses to 1.0)
    const float* W1 = (const float*)d_in[3];
    const float* b1 = (const float*)d_in[4];
    const float* W2 = (const float*)d_in[5];
    const float* b2 = (const float*)d_in[6];

    const int N = in_sizes[0] / IN;
    const int E = in_sizes[1] / 2;
    const int* src = ei;
    const int* dst = ei + E;

    // Workspace layout (floats):
    //   [0, N)                 deg -> dinv (in place)
    //   [N, N + N*128)         hw1 ; reused after conv1 as hw2 [0,N*64) + agg2 [N*64,N*128)
    //   [N + N*128, +N*128)    agg1 -> h2 (in place)
    float* ws   = (float*)d_ws;
    float* dinv = ws;
    float* hw1  = ws + N;
    float* h2   = hw1 + (size_t)N * HID;   // agg1, finalized in place to h2
    float* hw2  = hw1;                      // reuse region after finalize1
    float* agg2 = hw1 + (size_t)N * OUT;
    float* out  = (float*)d_out;

    const int zgrid = 2048;

    // ---- degree / normalization ----
    zero_kernel<<<zgrid, 256, 0, stream>>>(dinv, (size_t)N);
    zero_kernel<<<zgrid, 256, 0, stream>>>(h2, (size_t)N * HID);
    degree_kernel<<<(E + 255) / 256, 256, 0, stream>>>(dst, dinv, E);
    rsqrt_kernel<<<(N + 255) / 256, 256, 0, stream>>>(dinv, N);

    // ---- conv1: hw1 = x @ W1 (WMMA f32), scatter, epilogue ----
    gemm_wmma_kernel<IN, HID><<<(N + 127) / 128, 256, 0, stream>>>(x, W1, hw1, N);
    scatter_kernel<HID><<<(E + 7) / 8, 256, 0, stream>>>(src, dst, dinv, hw1, h2, E);
    finalize1_kernel<<<(N * HID + 255) / 256, 256, 0, stream>>>(h2, hw1, dinv, b1, N * HID);

    // ---- conv2: hw2 = h2 @ W2 (WMMA f32), scatter, log_softmax ----
    zero_kernel<<<zgrid, 256, 0, stream>>>(agg2, (size_t)N * OUT);
    gemm_wmma_kernel<HID, OUT><<<(N + 127) / 128, 256, 0, stream>>>(h2, W2, hw2, N);
    scatter_kernel<OUT><<<(E + 7) / 8, 256, 0, stream>>>(src, dst, dinv, hw2, agg2, E);
    finalize2_kernel<<<(N + 7) / 8, 256, 0, stream>>>(agg2, hw2, dinv, b2, out, N);
}